// ConvMHSA_1906965479829
// MI455X (gfx1250) — compile-verified
//
#include <hip/hip_runtime.h>

typedef __attribute__((ext_vector_type(2))) float v2f;
typedef __attribute__((ext_vector_type(4))) float v4f;
typedef __attribute__((ext_vector_type(8))) float v8f;
typedef __attribute__((ext_vector_type(4))) unsigned v4u;
typedef __attribute__((ext_vector_type(8))) unsigned v8u;

#define NB    8
#define TT    16
#define DIN   128
#define HH    32
#define WW    64
#define HW    2048
#define NHEAD 8
#define DK    32
#define DH    32
#define DOUT  128
#define BIMG  128      /* N*T */
#define CKLEN 65536    /* DK*HW */
#define HCLEN 256      /* NHEAD*DH */

__device__ __forceinline__ v8f wmma4(v2f a, v2f b, v8f c) {
  // D = A(16x4 f32) * B(4x16 f32) + C(16x16 f32), exact fp32 matrix op
  return __builtin_amdgcn_wmma_f32_16x16x4_f32(false, a, false, b, (short)0, c,
                                               false, false);
}

// ---------------------------------------------------------------------------
// Pack conv weights into A-fragment layout:
// wpack[ot(48)][tap(9)][cblk(32)][lane(32)][2]
//   lane<16: K = cblk*4 + {0,1};  lane>=16: K = cblk*4 + {2,3};  M = lane&15
// ---------------------------------------------------------------------------
__global__ void k_pack_conv(const float* __restrict__ qkw,
                            const float* __restrict__ vw,
                            float* __restrict__ wpack) {
  int tid = blockIdx.x * 256 + threadIdx.x;
  if (tid >= 48 * 9 * 32 * 32) return;
  int lane = tid & 31;
  int cblk = (tid >> 5) & 31;
  int tap  = (tid >> 10) % 9;
  int ot   = tid / 9216;
  int klo  = (lane < 16) ? 0 : 2;
  int o    = ot * 16 + (lane & 15);
  int cin0 = cblk * 4 + klo;
#pragma unroll
  for (int j = 0; j < 2; ++j) {
    int cin = cin0 + j;
    float val;
    if (o < 512) val = qkw[(o * DIN + cin) * 9 + tap];
    else         val = vw[((o - 512) * DIN + cin) * 9 + tap];
    wpack[tid * 2 + j] = val;
  }
}

// wproj[dt(8)][kb(64)][lane(32)][2] : A-fragment of W_out^T (M=d, K=hc)
__global__ void k_pack_proj(const float* __restrict__ Wout,
                            float* __restrict__ wproj) {
  int tid = blockIdx.x * 256 + threadIdx.x;
  if (tid >= 8 * 64 * 32) return;
  int lane = tid & 31;
  int kb   = (tid >> 5) & 63;
  int dt   = tid >> 11;
  int klo  = (lane < 16) ? 0 : 2;
  int d    = dt * 16 + (lane & 15);
#pragma unroll
  for (int j = 0; j < 2; ++j) {
    int hc = kb * 4 + klo + j;
    wproj[tid * 2 + j] = Wout[hc * DOUT + d];
  }
}

// ---------------------------------------------------------------------------
// Implicit-GEMM conv: one wave = 16 out-channels x 32 pixels (two N-tiles).
// 9 taps x 32 cin-blocks, K=4 fp32 WMMA each. x lives in L2 (134MB < 192MB).
// Writes q[n][h][t][c*HW+p], k likewise, v[b][hc][p].
// ---------------------------------------------------------------------------
__global__ void __launch_bounds__(256) k_conv(
    const float* __restrict__ x, const float* __restrict__ wpack,
    const float* __restrict__ qkb, const float* __restrict__ vb,
    float* __restrict__ qbuf, float* __restrict__ kbuf,
    float* __restrict__ vbuf) {
  int pp = blockIdx.x;              // 0..63 : 32-pixel pair-tile (one row half)
  int b  = blockIdx.y;              // 0..127 image = n*16+t
  int og = blockIdx.z;              // 0..5
  int wv   = threadIdx.x >> 5;
  int ot   = og * 8 + wv;           // 0..47 out-channel tile
  int lane = threadIdx.x & 31;
  int m    = lane & 15;
  int half = lane >> 4;
  int klo  = half * 2;

  int row     = pp >> 1;
  int colbase = (pp & 1) * 32;
  int c0      = colbase + m;
  const float* xb = x + (size_t)b * (DIN * HW);

  v8f acc0 = {};
  v8f acc1 = {};
  for (int tap = 0; tap < 9; ++tap) {
    int dy = tap / 3 - 1, dx = tap % 3 - 1;
    int r_  = row + dy;
    int cc0 = c0 + dx;
    int cc1 = cc0 + 16;
    bool vr = ((unsigned)r_)  < 32u;
    bool v0 = vr && ((unsigned)cc0) < 64u;
    bool v1 = vr && ((unsigned)cc1) < 64u;
    int pix0 = v0 ? (r_ * WW + cc0) : 0;   // clamped: loads always in-bounds
    int pix1 = v1 ? (r_ * WW + cc1) : 0;
    const float* wp = wpack + (size_t)(ot * 9 + tap) * 2048 + lane * 2;
    for (int cblk = 0; cblk < 32; ++cblk) {
      v2f a = *(const v2f*)(wp + cblk * 64);
      int cin = cblk * 4 + klo;
      const float* x0 = xb + (size_t)cin * HW;
      float t00 = x0[pix0];
      float t01 = x0[HW + pix0];
      float t10 = x0[pix1];
      float t11 = x0[HW + pix1];
      v2f b0 = {v0 ? t00 : 0.0f, v0 ? t01 : 0.0f};
      v2f b1 = {v1 ? t10 : 0.0f, v1 ? t11 : 0.0f};
      acc0 = wmma4(a, b0, acc0);
      acc1 = wmma4(a, b1, acc1);
    }
  }

  int n = b >> 4, t = b & 15;
  int p0 = row * WW + colbase + m;
  int p1 = p0 + 16;
#pragma unroll
  for (int i = 0; i < 8; ++i) {
    int o = ot * 16 + i + half * 8;       // D layout: M = i + 8*half, N = m
    float bias = (o < 512) ? qkb[o] : vb[o - 512];
    float val0 = acc0[i] + bias;
    float val1 = acc1[i] + bias;
    if (o < 512) {
      int h = o >> 6, s = (o >> 5) & 1, c = o & 31;   // o = h*64 + s*32 + c
      float* dst = s ? kbuf : qbuf;
      size_t base = ((size_t)((n * NHEAD + h) * TT + t)) * CKLEN + c * HW;
      dst[base + p0] = val0;
      dst[base + p1] = val1;
    } else {
      int hc = o - 512;
      size_t base = ((size_t)b * HCLEN + hc) * HW;
      vbuf[base + p0] = val0;
      vbuf[base + p1] = val1;
    }
  }
}

// ---------------------------------------------------------------------------
// Logits: A[nh][t][r] = sum_ck q[t][ck]*k[r][ck]. One wave per (nh, K-slice),
// 256 f32 WMMAs per wave; deterministic partials (no float atomics).
// Apart[slice(64)][nh(64)][256]
// ---------------------------------------------------------------------------
__global__ void __launch_bounds__(256) k_attn(const float* __restrict__ qbuf,
                                              const float* __restrict__ kbuf,
                                              float* __restrict__ Apart) {
  int nh    = blockIdx.x;            // 0..63
  int sg    = blockIdx.y;            // 0..7
  int wv    = threadIdx.x >> 5;
  int slice = sg * 8 + wv;           // 0..63
  int lane  = threadIdx.x & 31;
  int m     = lane & 15;
  int half  = lane >> 4;
  int klo   = half * 2;
  const float* qrow = qbuf + (size_t)nh * TT * CKLEN + (size_t)m * CKLEN;
  const float* krow = kbuf + (size_t)nh * TT * CKLEN + (size_t)m * CKLEN;
  int ck0 = slice * 1024 + klo;
  v8f acc = {};
  for (int it = 0; it < 256; ++it) {
    int ck = ck0 + it * 4;
    v2f a  = *(const v2f*)(qrow + ck);   // A: M=t,  K=ck
    v2f bb = *(const v2f*)(krow + ck);   // B: K=ck, N=r  (same addressing)
    acc = wmma4(a, bb, acc);
  }
  float* ap = Apart + ((size_t)slice * 64 + nh) * 256;
#pragma unroll
  for (int i = 0; i < 8; ++i) {
    int tt_ = i + half * 8;
    ap[tt_ * 16 + m] = acc[i];
  }
}

__global__ void k_reduceA(const float* __restrict__ Apart,
                          float* __restrict__ Abuf) {
  int nh  = blockIdx.x;     // 64
  int idx = threadIdx.x;    // 256
  float s = 0.0f;
  for (int sl = 0; sl < 64; ++sl)
    s += Apart[((size_t)sl * 64 + nh) * 256 + idx];
  Abuf[nh * 256 + idx] = s;
}

__global__ void k_softmax(const float* __restrict__ Abuf,
                          float* __restrict__ sAbuf) {
  int nh = blockIdx.x;
  int t  = threadIdx.x;
  if (t >= 16) return;
  const float* row = Abuf + nh * 256 + t * 16;
  float v_[16];
  float mx = -3.4e38f;
#pragma unroll
  for (int r = 0; r < 16; ++r) {
    v_[r] = row[r] * (1.0f / 256.0f);   // scale = sqrt(32*32*64) = 256
    mx = fmaxf(mx, v_[r]);
  }
  float s = 0.0f;
#pragma unroll
  for (int r = 0; r < 16; ++r) { v_[r] = expf(v_[r] - mx); s += v_[r]; }
  float inv = 1.0f / s;
  float* orow = sAbuf + nh * 256 + t * 16;
#pragma unroll
  for (int r = 0; r < 16; ++r) orow[r] = v_[r] * inv;
}

// ---------------------------------------------------------------------------
// Fused ctx + output projection. Per (n, 16-pixel tile): the Tensor Data
// Mover DMAs the 3D v tile (X=16 px contiguous, Y=256 ch stride 2048,
// Z=16 timesteps stride 524288) straight into LDS as vt[r][hc][p] (TDM fills
// X-fastest => exactly this layout), tracked on TENSORcnt. ctx is then built
// transposed [p][hc] (pitch 258: b64-aligned, bank-conflict-free) and the
// 256->128 projection runs as 64 K-chunk f32 WMMAs per (t, d-tile).
// ---------------------------------------------------------------------------
__global__ void __launch_bounds__(512) k_ctx_proj(
    const float* __restrict__ vbuf, const float* __restrict__ sAbuf,
    const float* __restrict__ wproj, const float* __restrict__ bout,
    float* __restrict__ out) {
  extern __shared__ float smem[];
  float* vt  = smem;                        // 65536 : vt[r][hc][p] (LDS off 0)
  float* ctx = smem + 65536;                // 2 * 4128 : ctx[tsel][p*258+hc]
  float* sAl = smem + 65536 + 2 * 4128;     // 2048 : sA[h][t][r] for this n

  int pt = blockIdx.x;          // 0..127 pixel tile
  int n  = blockIdx.y;          // 0..7
  int pbase = pt * 16;
  int tid  = threadIdx.x;
  int lane = tid & 31;
  int wv   = tid >> 5;
  int m    = lane & 15;
  int half = lane >> 4;
  int klo  = half * 2;

#pragma unroll
  for (int e = 0; e < 4; ++e) {
    int i = tid + e * 512;
    sAl[i] = sAbuf[n * 2048 + i];
  }

  if (wv == 0) {
    // ---- Tensor DMA descriptor (D#), CDNA5 ISA ----
    unsigned long long gaddr = (unsigned long long)(const void*)(
        vbuf + (size_t)n * TT * HCLEN * HW + pbase);
    // group0: count=1 | lds_addr=0 | global_addr | type=2 (bits 127:126)
    v4u g0 = {1u, 0u, (unsigned)gaddr,
              (unsigned)(gaddr >> 32) | 0x80000000u};
    // group1: data_size=4B(code 2); tensor_dim0=2048, tensor_dim1=256;
    //         tile_dim0=16, tile_dim1=256, tile_dim2=16;
    //         dim0_stride=2048, dim1_stride=524288 (elements)
    v8u g1 = {0x00020000u,   // mask=0, data_size=2
              0x08000000u,   // abar=0 | tensor_dim0.lo16=2048
              0x01000000u,   // tensor_dim0.hi16=0 | tensor_dim1.lo16=256
              0x00100000u,   // tensor_dim1.hi16=0 | tile_dim0=16
              0x00100100u,   // tile_dim1=256 | tile_dim2=16
              2048u,         // tensor_dim0_stride[31:0]
              0u,            // d0_stride[47:32]=0 | d1_stride[15:0]=0
              8u};           // tensor_dim1_stride[47:16] = 0x80000>>16
    // group2: tensor_dim2=16; tensor_dim2_stride=524288; tile_dim3=0
    v4u g2 = {16u, 0u, 524288u, 0u};
    v4u g3 = {0u, 0u, 0u, 0u};
    asm volatile("tensor_load_to_lds %0, %1, %2, %3"
                 :
                 : "s"(g0), "s"(g1), "s"(g2), "s"(g3)
                 : "memory");
    __builtin_amdgcn_s_wait_tensorcnt(0);
  }
  __syncthreads();   // publish TDM-filled vt to all waves

  int tsel_w = wv >> 3;         // which of the two t's this wave projects
  int dt     = wv & 7;          // d-tile 0..7

  for (int tp = 0; tp < 8; ++tp) {
    // ctx for t = 2*tp + {0,1}
#pragma unroll
    for (int e = 0; e < 4; ++e) {
      int q    = tid + e * 512;             // 0..2047 quad id
      int tsel = q >> 10;
      int rem  = q & 1023;
      int hc   = rem >> 2;
      int p4   = rem & 3;
      int t    = tp * 2 + tsel;
      const float* sa = sAl + (hc >> 5) * 256 + t * 16;
      float sx = 0.f, sy = 0.f, sz = 0.f, sw = 0.f;
#pragma unroll
      for (int r = 0; r < 16; ++r) {
        v4f vv = *(const v4f*)(vt + r * 4096 + hc * 16 + p4 * 4);
        float w_ = sa[r];
        sx += w_ * vv.x; sy += w_ * vv.y; sz += w_ * vv.z; sw += w_ * vv.w;
      }
      float* cbase = ctx + tsel * 4128 + hc;
      cbase[(p4 * 4 + 0) * 258] = sx;
      cbase[(p4 * 4 + 1) * 258] = sy;
      cbase[(p4 * 4 + 2) * 258] = sz;
      cbase[(p4 * 4 + 3) * 258] = sw;
    }
    __syncthreads();

    // projection: out[d][p] = sum_hc W_out[hc][d] * ctx[hc][p]
    int t = tp * 2 + tsel_w;
    v8f acc = {};
    const float* wpd = wproj + (size_t)dt * 4096 + lane * 2;
    const float* cb  = ctx + tsel_w * 4128 + m * 258 + klo;
    for (int kb = 0; kb < 64; ++kb) {
      v2f a  = *(const v2f*)(wpd + kb * 64);  // A: M=d,  K=hc
      v2f bb = *(const v2f*)(cb + kb * 4);    // B: K=hc, N=p (LDS b64)
      acc = wmma4(a, bb, acc);
    }
    size_t obase = (size_t)(n * TT + t) * DOUT * HW + pbase + m;
#pragma unroll
    for (int i = 0; i < 8; ++i) {
      int d = dt * 16 + i + half * 8;
      out[obase + (size_t)d * HW] = acc[i] + bout[d];
    }
    __syncthreads();
  }
}

// ---------------------------------------------------------------------------
extern "C" void kernel_launch(void* const* d_in, const int* in_sizes, int n_in,
                              void* d_out, int out_size, void* d_ws,
                              size_t ws_size, hipStream_t stream) {
  const float* x    = (const float*)d_in[0];
  const float* qkw  = (const float*)d_in[1];
  const float* qkb  = (const float*)d_in[2];
  const float* vw   = (const float*)d_in[3];
  const float* vb   = (const float*)d_in[4];
  const float* Wout = (const float*)d_in[5];
  const float* bout = (const float*)d_in[6];
  float* out = (float*)d_out;
  float* ws  = (float*)d_ws;

  const size_t QS = 67108864ull;          // 8*8*16*65536 floats
  float* qbuf  = ws;                      // 268 MB
  float* kbuf  = ws + QS;                 // 268 MB
  float* vbuf  = ws + 2 * QS;             // 268 MB  (128*256*2048)
  float* Abuf  = ws + 3 * QS;             // 16384
  float* sAbuf = Abuf + 16384;            // 16384
  float* Apart = sAbuf + 16384;           // 64*64*256 = 1048576
  float* wpack = Apart + 1048576;         // 884736
  float* wproj = wpack + 884736;          // 32768
  // total ~203.3M floats (~813 MB) in d_ws

  k_pack_conv<<<1728, 256, 0, stream>>>(qkw, vw, wpack);
  k_pack_proj<<<64, 256, 0, stream>>>(Wout, wproj);
  k_conv<<<dim3(64, 128, 6), 256, 0, stream>>>(x, wpack, qkb, vb, qbuf, kbuf,
                                               vbuf);
  k_attn<<<dim3(64, 8), 256, 0, stream>>>(qbuf, kbuf, Apart);
  k_reduceA<<<64, 256, 0, stream>>>(Apart, Abuf);
  k_softmax<<<64, 32, 0, stream>>>(Abuf, sAbuf);
  k_ctx_proj<<<dim3(128, 8), 512, 303360, stream>>>(vbuf, sAbuf, wproj, bout,
                                                    out);
}